// MultiHeadAttention_81149112090633
// MI455X (gfx1250) — compile-verified
//
#include <hip/hip_runtime.h>

// ---------------------------------------------------------------------------
// MultiHeadAttention (faithful to reference quirks) for MI455X / gfx1250.
// All matmuls via V_WMMA_F32_16X16X4_F32 (wave32 WMMA, fp32 = reference dtype).
// Output-projection GEMM uses double-buffered GLOBAL_LOAD_ASYNC_TO_LDS_B128
// staging with s_wait_asynccnt (CDNA5 async copy path).
// ---------------------------------------------------------------------------

typedef __attribute__((ext_vector_type(2))) float v2f;
typedef __attribute__((ext_vector_type(8))) float v8f;

#define NB    256        // batch
#define LSEQ  64         // sequence length
#define NH    64         // heads
#define HD    64         // head dim
#define EMB   4096       // embed
#define MROWS (NB * NH)  // 16384 rows of `pre`

static __device__ __forceinline__ v8f v8f_zero() {
  v8f z = {0.f, 0.f, 0.f, 0.f, 0.f, 0.f, 0.f, 0.f};
  return z;
}

// D = A(16x4 f32) * B(4x16 f32) + C(16x16 f32), one wave, wave32.
static __device__ __forceinline__ v8f wmma_f32(v2f a, v2f b, v8f c) {
  return __builtin_amdgcn_wmma_f32_16x16x4_f32(
      /*neg_a=*/false, a, /*neg_b=*/false, b,
      /*c_mod=*/(short)0, c, /*reuse_a=*/false, /*reuse_b=*/false);
}

// Async copy 16 bytes global -> LDS (no VGPR round-trip, tracked by ASYNCcnt).
// Generic shared address truncated to 32 bits == wave-relative LDS offset
// (ISA 10.2: LDS aperture address maps via addr[31:0]).
static __device__ __forceinline__ void async_copy_b128(void* lds,
                                                       const void* gptr) {
  unsigned lds_off = (unsigned)(unsigned long long)lds;
  asm volatile("global_load_async_to_lds_b128 %0, %1, off"
               :
               : "v"(lds_off), "v"(gptr)
               : "memory");
}

// ---------------------------------------------------------------------------
// Kernel 1: Vsum[n,l,d] = sum_v values[n,l, v*64 + d]   (collapses the 'v' sum)
// grid = (NB*LSEQ), block = 64
// ---------------------------------------------------------------------------
__global__ void vsum_kernel(const float* __restrict__ values,
                            float* __restrict__ vsum) {
  const int row = blockIdx.x;   // n*64 + l
  const int d   = threadIdx.x;  // 0..63
  const float* src = values + (size_t)row * EMB + d;
  float acc = 0.f;
#pragma unroll
  for (int v = 0; v < NH; ++v) acc += src[v * HD];
  vsum[(size_t)row * HD + d] = acc;
}

// ---------------------------------------------------------------------------
// Kernel 2: per (l, n-chunk): E = Q_{n,l} @ K_{n,l}^T (64x64x64 via WMMA),
// P = exp(E/8) (masked n skipped -> contributes 0), S[l,q,k] += sum_n P.
// grid = (LSEQ, 8), block = 256 (8 waves; 2 of 16 output tiles per wave)
// ---------------------------------------------------------------------------
__global__ __launch_bounds__(256) void energy_exp_kernel(
    const float* __restrict__ query, const float* __restrict__ keys,
    const int* __restrict__ mask, float* __restrict__ P,
    float* __restrict__ S) {
  __shared__ float Qs[64 * 65];
  __shared__ float Ks[64 * 65];
  const int l    = blockIdx.x;
  const int n0   = blockIdx.y * 32;
  const int tid  = threadIdx.x;
  const int lane = tid & 31;
  const int wave = tid >> 5;
  const int lr   = lane & 15;        // row/col within 16-wide tile
  const int ds   = (lane >> 4) * 2;  // K offset (0 or 2) per ISA A/B layout

  v8f sacc[2];
  sacc[0] = v8f_zero();
  sacc[1] = v8f_zero();

  for (int n = n0; n < n0 + 32; ++n) {
    if (mask[n * LSEQ + l] == 0) continue;  // uniform across block
    __syncthreads();
    const size_t rowbase = ((size_t)n * LSEQ + l) * EMB;
    for (int idx = tid; idx < 4096; idx += 256) {
      const int r = idx >> 6, c = idx & 63;
      Qs[r * 65 + c] = query[rowbase + idx];  // Qmat[q][d]
      Ks[r * 65 + c] = keys[rowbase + idx];   // Kmat[k][d]
    }
    __syncthreads();
#pragma unroll
    for (int t = 0; t < 2; ++t) {
      const int tile = wave * 2 + t;
      const int qi = tile >> 2, ki = tile & 3;
      v8f e = v8f_zero();
#pragma unroll
      for (int kk = 0; kk < 16; ++kk) {  // contraction over d in steps of 4
        v2f a, b;
        a.x = Qs[(qi * 16 + lr) * 65 + kk * 4 + ds];
        a.y = Qs[(qi * 16 + lr) * 65 + kk * 4 + ds + 1];
        b.x = Ks[(ki * 16 + lr) * 65 + kk * 4 + ds];
        b.y = Ks[(ki * 16 + lr) * 65 + kk * 4 + ds + 1];
        e = wmma_f32(a, b, e);
      }
#pragma unroll
      for (int r = 0; r < 8; ++r) {  // C layout: VGPR r -> M=r (+8 hi half)
        const int q  = qi * 16 + r + ((lane >> 4) << 3);
        const int kc = ki * 16 + lr;
        const float p = __expf(e[r] * 0.125f);  // scale 1/sqrt(64)
        P[(((size_t)n * LSEQ + l) * 64 + q) * 64 + kc] = p;
        sacc[t][r] += p;
      }
    }
  }
  // flush softmax denominators (8 n-chunks accumulate -> atomic)
#pragma unroll
  for (int t = 0; t < 2; ++t) {
    const int tile = wave * 2 + t;
    const int qi = tile >> 2, ki = tile & 3;
#pragma unroll
    for (int r = 0; r < 8; ++r) {
      const int q  = qi * 16 + r + ((lane >> 4) << 3);
      const int kc = ki * 16 + lr;
      atomicAdd(&S[((size_t)l * 64 + q) * 64 + kc], sacc[t][r]);
    }
  }
}

// ---------------------------------------------------------------------------
// Kernel 3: per (n,k): A[q,l] = P/S (0 when masked), out = A @ Vsum[n]
// (64x64x64 via WMMA). Writes pre[k*256+n, q*64+d] (one full row of `pre`).
// grid = (NB, NH), block = 256
// ---------------------------------------------------------------------------
__global__ __launch_bounds__(256) void attn_out_kernel(
    const float* __restrict__ P, const float* __restrict__ S,
    const float* __restrict__ vsum, const int* __restrict__ mask,
    float* __restrict__ pre) {
  __shared__ float As[64 * 65];  // A[q][l]
  __shared__ float Vs[64 * 65];  // Vsum[l][d]
  const int n    = blockIdx.x;
  const int k    = blockIdx.y;
  const int tid  = threadIdx.x;
  const int lane = tid & 31;
  const int wave = tid >> 5;
  const int lr   = lane & 15;
  const int ds   = (lane >> 4) * 2;

  for (int idx = tid; idx < 4096; idx += 256) {
    const int c  = idx & 63;  // q for As, d for Vs
    const int li = idx >> 6;
    float a = 0.f;
    if (mask[n * LSEQ + li] != 0) {
      const float num = P[(((size_t)n * LSEQ + li) * 64 + c) * 64 + k];
      const float den = S[((size_t)li * 64 + c) * 64 + k];
      a = num / den;
    }
    As[c * 65 + li] = a;
    Vs[li * 65 + c] = vsum[((size_t)n * LSEQ + li) * HD + c];
  }
  __syncthreads();

  const size_t m = (size_t)k * NB + n;  // pre row (derived from reshape quirk)
#pragma unroll
  for (int t = 0; t < 2; ++t) {
    const int tile = wave * 2 + t;
    const int qi = tile >> 2, di = tile & 3;
    v8f c = v8f_zero();
#pragma unroll
    for (int kk = 0; kk < 16; ++kk) {  // contraction over l
      v2f a, b;
      a.x = As[(qi * 16 + lr) * 65 + kk * 4 + ds];
      a.y = As[(qi * 16 + lr) * 65 + kk * 4 + ds + 1];
      b.x = Vs[(kk * 4 + ds) * 65 + di * 16 + lr];
      b.y = Vs[(kk * 4 + ds + 1) * 65 + di * 16 + lr];
      c = wmma_f32(a, b, c);
    }
#pragma unroll
    for (int r = 0; r < 8; ++r) {
      const int q = qi * 16 + r + ((lane >> 4) << 3);
      const int d = di * 16 + lr;
      pre[m * EMB + q * 64 + d] = c[r];
    }
  }
}

// ---------------------------------------------------------------------------
// Kernel 4: out = pre(16384x4096) @ w_out^T(4096x4096) + b_out.
// 128x128 block tile, 8 waves x (16x128) strips. K staged 16-wide in LDS via
// double-buffered async global->LDS B128 copies (ASYNCcnt pipelined: the
// next chunk's copies are in flight while the current chunk feeds WMMA).
// LDS row stride 20 floats: 16B-aligned for B128 and bank-conflict-free.
// grid = (MROWS/128, EMB/128), block = 256
// ---------------------------------------------------------------------------
#define STR4 20  // LDS row stride in floats

__global__ __launch_bounds__(256) void out_gemm_kernel(
    const float* __restrict__ pre, const float* __restrict__ w_out,
    const float* __restrict__ b_out, float* __restrict__ out) {
  __shared__ float As[2][128 * STR4];  // pre  rows (M) x K-chunk
  __shared__ float Bs[2][128 * STR4];  // w_out rows (N) x K-chunk (B^T staging)
  const int bm   = blockIdx.x;
  const int bn   = blockIdx.y;
  const int tid  = threadIdx.x;
  const int lane = tid & 31;
  const int wave = tid >> 5;
  const int lr   = lane & 15;
  const int ds   = (lane >> 4) * 2;

  v8f c[8];
#pragma unroll
  for (int j = 0; j < 8; ++j) c[j] = v8f_zero();

  // Issue the async staging of one 16-wide K chunk (4 async instrs per wave).
  auto issue_chunk = [&](int buf, int ke) {
#pragma unroll
    for (int s = 0; s < 4; ++s) {
      const int i  = s * 256 + tid;  // 0..1023: 512 f4 for As, 512 for Bs
      const int r  = (i >> 2) & 127;
      const int c4 = (i & 3) * 4;
      if (i < 512) {
        async_copy_b128(&As[buf][r * STR4 + c4],
                        pre + (size_t)(bm * 128 + r) * EMB + ke + c4);
      } else {
        async_copy_b128(&Bs[buf][r * STR4 + c4],
                        w_out + (size_t)(bn * 128 + r) * EMB + ke + c4);
      }
    }
  };

  issue_chunk(0, 0);
  for (int ch = 0; ch < EMB / 16; ++ch) {
    const int cur = ch & 1;
    if (ch + 1 < EMB / 16) {
      issue_chunk(cur ^ 1, (ch + 1) * 16);
      // wait until only the 4 newest (next-chunk) async copies are in flight
      asm volatile("s_wait_asynccnt 4" ::: "memory");
    } else {
      asm volatile("s_wait_asynccnt 0" ::: "memory");
    }
    __syncthreads();  // all waves' copies of `cur` landed in LDS
    const float* Ab = As[cur];
    const float* Bb = Bs[cur];
#pragma unroll
    for (int kk = 0; kk < 4; ++kk) {
      v2f a;  // shared across all 8 N-tiles of this wave
      a.x = Ab[(wave * 16 + lr) * STR4 + kk * 4 + ds];
      a.y = Ab[(wave * 16 + lr) * STR4 + kk * 4 + ds + 1];
#pragma unroll
      for (int ji = 0; ji < 8; ++ji) {
        v2f b;
        b.x = Bb[(ji * 16 + lr) * STR4 + kk * 4 + ds];
        b.y = Bb[(ji * 16 + lr) * STR4 + kk * 4 + ds + 1];
        c[ji] = wmma_f32(a, b, c[ji]);
      }
    }
    __syncthreads();  // compute done before buffer `cur` is refilled
  }

#pragma unroll
  for (int ji = 0; ji < 8; ++ji) {
#pragma unroll
    for (int r = 0; r < 8; ++r) {
      const int row = bm * 128 + wave * 16 + r + ((lane >> 4) << 3);
      const int col = bn * 128 + ji * 16 + lr;
      out[(size_t)row * EMB + col] = c[ji][r] + b_out[col];
    }
  }
}

// ---------------------------------------------------------------------------
// Workspace layout (floats):
//   vsum : 1,048,576   (4 MB)
//   S    :   262,144   (1 MB)   -- softmax denominators, zeroed each launch
//   P    : 67,108,864  (256 MB) -- exp(energy/8)
//   pre  : 67,108,864  (256 MB) -- projection input
// total ~517 MB of d_ws.
// ---------------------------------------------------------------------------
extern "C" void kernel_launch(void* const* d_in, const int* in_sizes, int n_in,
                              void* d_out, int out_size, void* d_ws,
                              size_t ws_size, hipStream_t stream) {
  (void)in_sizes; (void)n_in; (void)out_size; (void)ws_size;
  const float* values = (const float*)d_in[0];
  const float* keys   = (const float*)d_in[1];
  const float* query  = (const float*)d_in[2];
  const int*   mask   = (const int*)d_in[3];
  const float* w_out  = (const float*)d_in[4];
  const float* b_out  = (const float*)d_in[5];
  float* out = (float*)d_out;

  float* ws   = (float*)d_ws;
  float* vsum = ws;
  float* S    = ws + (size_t)1048576;
  float* P    = ws + (size_t)1048576 + 262144;
  float* pre  = P + (size_t)67108864;

  hipMemsetAsync(S, 0, 262144 * sizeof(float), stream);  // capture-safe
  vsum_kernel<<<dim3(NB * LSEQ), dim3(64), 0, stream>>>(values, vsum);
  energy_exp_kernel<<<dim3(LSEQ, 8), dim3(256), 0, stream>>>(query, keys, mask,
                                                             P, S);
  attn_out_kernel<<<dim3(NB, NH), dim3(256), 0, stream>>>(P, S, vsum, mask,
                                                          pre);
  out_gemm_kernel<<<dim3(MROWS / 128, EMB / 128), dim3(256), 0, stream>>>(
      pre, w_out, b_out, out);
}